// FourierCrossAttention_60455959658506
// MI455X (gfx1250) — compile-verified
//
#include <hip/hip_runtime.h>
#include <math.h>

typedef __attribute__((ext_vector_type(2))) float v2f;
typedef __attribute__((ext_vector_type(8))) float v8f;

#define Bn 16
#define Ln 2048
#define Hn 8
#define En 64
#define NF 64   // kept modes

// ---- workspace layout (float offsets) ----
#define OFF_COS 0
#define OFF_SIN (NF * Ln)                  // 131072
#define OFF_XQR (2 * NF * Ln)              // 262144
#define SZ_X    (Bn * Hn * 64 * 64)        // 524288
#define OFF_XQI (OFF_XQR + SZ_X)
#define OFF_XKR (OFF_XQI + SZ_X)
#define OFF_XKI (OFF_XKR + SZ_X)
#define OFF_TR  (OFF_XKI + SZ_X)
#define OFF_TI  (OFF_TR + SZ_X)
#define OFF_UR  (OFF_TI + SZ_X)
#define OFF_UI  (OFF_UR + SZ_X)
// total = 262144 + 8*524288 = 4456448 floats (~17.8 MB)

// ---- fp32 WMMA 16x16x4 wrappers (wave32) ----
__device__ __forceinline__ v8f wmma4(v2f a, v2f b, v8f c) {
  return __builtin_amdgcn_wmma_f32_16x16x4_f32(false, a, false, b, (short)0, c,
                                               false, false);
}
// A fragment: 16x4 tile, row-major [m][k], lanes 0-15: M=lane,K={0,1}; 16-31: K={2,3}
__device__ __forceinline__ v2f load_a(const float* A, int lda, int lane, float sgn) {
  int m = lane & 15, k = (lane >> 4) << 1;
  v2f r; r.x = sgn * A[m * lda + k]; r.y = sgn * A[m * lda + k + 1]; return r;
}
// B fragment: 4x16 tile, row-major [k][n]
__device__ __forceinline__ v2f load_b(const float* Bm, int ldb, int lane) {
  int n = lane & 15, k = (lane >> 4) << 1;
  v2f r; r.x = Bm[k * ldb + n]; r.y = Bm[(k + 1) * ldb + n]; return r;
}
// B fragment from a transposed (column-major) source: B[k][n] = Mt[n*ld + k]
__device__ __forceinline__ v2f load_bt(const float* Mt, int ld, int lane) {
  int n = lane & 15, k = (lane >> 4) << 1;
  v2f r; r.x = Mt[n * ld + k]; r.y = Mt[n * ld + k + 1]; return r;
}

// ============================================================================
// Kernel 1: DFT basis tables.  cos/sin(2*pi*f*t/L), exact period reduction.
// rfft: X[f] = sum_t x[t] cos - i * sum_t x[t] sin
// ============================================================================
__global__ void k_basis(float* __restrict__ ws) {
  int idx = blockIdx.x * 256 + threadIdx.x;
  if (idx >= NF * Ln) return;
  int f = idx >> 11, t = idx & (Ln - 1);
  int r = (f * t) & (Ln - 1);                       // exact mod-L reduction
  float th = (float)r * 0.0030679615757712823f;     // 2*pi/2048
  ws[OFF_COS + idx] = cosf(th);
  ws[OFF_SIN + idx] = sinf(th);
}

// ============================================================================
// Kernel 2: mode-truncated DFT + frequency-domain projection, per (b,h,{q|k}).
//   phase 1: F[f][j](re,im) = sum_t basis[f][t] * x[b,t,h,j]     (K=2048)
//   phase 2: X[f][e]        = sum_j F[f][j] * W[e][j]  (+ L*bias on f==0 real)
// 8 waves; each wave owns 4 of 32 16x16 tiles (part re/im, tf, 4 tj/te).
// ============================================================================
__global__ __launch_bounds__(256) void k_dft_proj(
    const float* __restrict__ q, const float* __restrict__ k,
    const float* __restrict__ wq_w, const float* __restrict__ wq_b,
    const float* __restrict__ wk_w, const float* __restrict__ wk_b,
    float* __restrict__ ws) {
  // staging: x-chunk [32][64] @0 ; cos [64][33] @2048 ; sin [64][33] @4160
  // phase 2 reuses [0..4159] as W[64][65]
  __shared__ float stag[6272];
  __shared__ float ldsF[2][64][65];   // DFT result (re,im), padded stride

  int bx = blockIdx.x;                // ((b*Hn + h)*2 + s)
  int s = bx & 1, bh = bx >> 1;
  int h = bh & (Hn - 1), b = bh >> 3;
  int tid = threadIdx.x, lane = tid & 31, wv = tid >> 5;
  int part = wv >> 2;                 // 0 = real, 1 = imag
  int tf = (wv & 3) << 4;

  const float* x    = s ? k : q;
  const float* W    = s ? wk_w : wq_w;
  const float* bias = s ? wk_b : wq_b;

  // ---- phase 1: F = basis * x  (contraction over t, chunks of 32) ----
  v8f acc[4] = {};
  for (int kc = 0; kc < Ln / 32; ++kc) {
    int t0 = kc * 32;
    for (int i = tid; i < 2048; i += 256) {         // x chunk [t][j]
      int tt = i >> 6, j = i & 63;
      stag[i] = x[((size_t)((b * Ln + t0 + tt) * Hn + h)) * En + j];
    }
    for (int i = tid; i < 2048; i += 256) {         // basis chunk [f][t]
      int f = i >> 5, tt = i & 31;
      stag[2048 + f * 33 + tt] = ws[OFF_COS + f * Ln + t0 + tt];
      stag[4160 + f * 33 + tt] = ws[OFF_SIN + f * Ln + t0 + tt];
    }
    __syncthreads();
    const float* Ab = part ? (stag + 4160) : (stag + 2048);
    for (int k0 = 0; k0 < 32; k0 += 4) {
      v2f a = load_a(Ab + tf * 33 + k0, 33, lane, 1.0f);
#pragma unroll
      for (int i = 0; i < 4; ++i) {
        v2f bb = load_b(stag + k0 * 64 + i * 16, 64, lane);
        acc[i] = wmma4(a, bb, acc[i]);
      }
    }
    __syncthreads();
  }
  {   // write F to LDS; imag part is -sum(x*sin)
    int n = lane & 15, mh = (lane >> 4) * 8;
    float sg = part ? -1.0f : 1.0f;
    for (int i = 0; i < 4; ++i)
      for (int r = 0; r < 8; ++r)
        ldsF[part][tf + mh + r][i * 16 + n] = sg * acc[i][r];
  }
  __syncthreads();

  // ---- phase 2: X[f][e] = sum_j F[f][j] * W[e][j] ----
  for (int i = tid; i < 4096; i += 256)             // W[e][j] -> [64][65]
    stag[(i >> 6) * 65 + (i & 63)] = W[i];
  __syncthreads();

  v8f acc2[4] = {};
  for (int k0 = 0; k0 < 64; k0 += 4) {
    v2f a = load_a(&ldsF[part][tf][k0], 65, lane, 1.0f);
#pragma unroll
    for (int i = 0; i < 4; ++i) {
      v2f bb = load_bt(stag + (i * 16) * 65 + k0, 65, lane);  // B[j][e]=W[e][j]
      acc2[i] = wmma4(a, bb, acc2[i]);
    }
  }
  float* Xout = ws + (s ? (part ? OFF_XKI : OFF_XKR) : (part ? OFF_XQI : OFF_XQR))
              + (size_t)bh * 4096;
  {
    int n = lane & 15, mh = (lane >> 4) * 8;
    for (int i = 0; i < 4; ++i)
      for (int r = 0; r < 8; ++r)
        Xout[(tf + mh + r) * 64 + i * 16 + n] = acc2[i][r];
  }
  __syncthreads();
  // bias only reaches DC bin: X[0][e] += L * bias[e] (real part)
  if (tid < 64) {
    float* Xr = ws + (s ? OFF_XKR : OFF_XQR) + (size_t)bh * 4096;
    Xr[tid] += 2048.0f * bias[tid];
  }
}

// ============================================================================
// Kernel 3: per (b,h): S = Xq^T Xk (complex, over e), tanh(re/im),
//           T[x][e] = sum_y S[x,y] * Xk[e,y]  (complex, over y)
// Xk spectrum and S live in LDS (exactly 64 KB).  T stored [b][h][x][e].
// ============================================================================
__global__ __launch_bounds__(256) void k_attn(float* __restrict__ ws) {
  __shared__ float Kre[64][64];
  __shared__ float Kim[64][64];
  __shared__ float Sre[64][64];
  __shared__ float Sim[64][64];

  int bh = blockIdx.x;
  int tid = threadIdx.x, lane = tid & 31, wv = tid >> 5;
  int part = wv >> 2;                 // 0 = real, 1 = imag
  int tx = (wv & 3) << 4;
  size_t xo = (size_t)bh * 4096;
  const float* XqR = ws + OFF_XQR + xo;
  const float* XqI = ws + OFF_XQI + xo;
  const float* XkR = ws + OFF_XKR + xo;
  const float* XkI = ws + OFF_XKI + xo;

  for (int i = tid; i < 4096; i += 256) {
    Kre[i >> 6][i & 63] = XkR[i];
    Kim[i >> 6][i & 63] = XkI[i];
  }
  __syncthreads();

  // ---- S[x][y] = sum_e Xq[e,x]*Xk[e,y]  (stored layouts are [f][e]) ----
  v8f acc[4] = {};
  for (int k0 = 0; k0 < 64; k0 += 4) {
    v2f aqr = load_a(XqR + tx * 64 + k0, 64, lane, 1.0f);
    v2f aqi = load_a(XqI + tx * 64 + k0, 64, lane, part ? 1.0f : -1.0f);
#pragma unroll
    for (int i = 0; i < 4; ++i) {
      int ty = i << 4;
      v2f bkr = load_bt(&Kre[ty][k0], 64, lane);   // B[e][y] = XkT[y][e]
      v2f bki = load_bt(&Kim[ty][k0], 64, lane);
      if (part == 0) {                // Sr = Qr*Kr - Qi*Ki
        acc[i] = wmma4(aqr, bkr, acc[i]);
        acc[i] = wmma4(aqi, bki, acc[i]);
      } else {                        // Si = Qr*Ki + Qi*Kr
        acc[i] = wmma4(aqr, bki, acc[i]);
        acc[i] = wmma4(aqi, bkr, acc[i]);
      }
    }
  }
  {
    int n = lane & 15, mh = (lane >> 4) * 8;
    float (*Sd)[64] = part ? Sim : Sre;
    for (int i = 0; i < 4; ++i)
      for (int r = 0; r < 8; ++r)
        Sd[tx + mh + r][i * 16 + n] = acc[i][r];
  }
  __syncthreads();
  for (int i = tid; i < 8192; i += 256) {           // tanh on re and im
    float* p = (i < 4096) ? &Sre[i >> 6][i & 63] : &Sim[(i - 4096) >> 6][i & 63];
    *p = tanhf(*p);
  }
  __syncthreads();

  // ---- T[x][e] = sum_y S[x,y] * XkT[y][e] ----
  v8f acc2[4] = {};
  for (int k0 = 0; k0 < 64; k0 += 4) {
    v2f asr = load_a(&Sre[tx][k0], 64, lane, 1.0f);
    v2f asi = load_a(&Sim[tx][k0], 64, lane, part ? 1.0f : -1.0f);
#pragma unroll
    for (int i = 0; i < 4; ++i) {
      int te = i << 4;
      v2f bkr = load_b(&Kre[k0][te], 64, lane);
      v2f bki = load_b(&Kim[k0][te], 64, lane);
      if (part == 0) {                // Tr = Sr*Kr - Si*Ki
        acc2[i] = wmma4(asr, bkr, acc2[i]);
        acc2[i] = wmma4(asi, bki, acc2[i]);
      } else {                        // Ti = Sr*Ki + Si*Kr
        acc2[i] = wmma4(asr, bki, acc2[i]);
        acc2[i] = wmma4(asi, bkr, acc2[i]);
      }
    }
  }
  float* Tout = ws + (part ? OFF_TI : OFF_TR) + (size_t)bh * 4096;
  {
    int n = lane & 15, mh = (lane >> 4) * 8;
    for (int i = 0; i < 4; ++i)
      for (int r = 0; r < 8; ++r)
        Tout[(tx + mh + r) * 64 + i * 16 + n] = acc2[i][r];
  }
}

// ============================================================================
// Kernel 4: per (h,x): U[b,o] = sum_e T[b,e] * (w1 + i w2)[h,e,o,x]
// M=16 (batch == one tile), N=64, K=64.  8 waves = {re,im} x 4 o-tiles.
// U stored [b][h][o][x].
// ============================================================================
__global__ __launch_bounds__(256) void k_modemix(
    const float* __restrict__ w1, const float* __restrict__ w2,
    float* __restrict__ ws) {
  int bx = blockIdx.x;                // h*64 + x
  int h = bx >> 6, xm = bx & 63;
  int tid = threadIdx.x, lane = tid & 31, wv = tid >> 5;
  int part = wv >> 2;
  int to = (wv & 3) << 4;
  const int ldT = Hn * 64 * 64;       // batch stride in T
  const float* TR = ws + OFF_TR + (size_t)(h * 64 + xm) * 64;
  const float* TI = ws + OFF_TI + (size_t)(h * 64 + xm) * 64;

  v8f acc = {};
  for (int k0 = 0; k0 < 64; k0 += 4) {
    v2f atr = load_a(TR + k0, ldT, lane, 1.0f);
    v2f ati = load_a(TI + k0, ldT, lane, part ? 1.0f : -1.0f);
    int n = lane & 15, kk = (lane >> 4) << 1;
    size_t wo = ((size_t)((h * 64 + k0 + kk) * 64 + to + n)) * 64 + xm;
    v2f bwr; bwr.x = w1[wo]; bwr.y = w1[wo + 4096];   // next e row: +64*64
    v2f bwi; bwi.x = w2[wo]; bwi.y = w2[wo + 4096];
    if (part == 0) {                  // Ur = Tr*W1 - Ti*W2
      acc = wmma4(atr, bwr, acc);
      acc = wmma4(ati, bwi, acc);
    } else {                          // Ui = Tr*W2 + Ti*W1
      acc = wmma4(atr, bwi, acc);
      acc = wmma4(ati, bwr, acc);
    }
  }
  float* U = ws + (part ? OFF_UI : OFF_UR);
  int n = lane & 15, mh = (lane >> 4) * 8;
  for (int r = 0; r < 8; ++r) {
    int b = mh + r;
    U[((size_t)(b * Hn + h) * 64 + to + n) * 64 + xm] = acc[r];
  }
}

// ============================================================================
// Kernel 5: inverse rDFT with 64-mode spectrum, as GEMM [64o x 128k][128k x 2048t]
//   out[t] = s*(Ur[0] + 2*sum_f (Ur[f] cos - Ui[f] sin)),  s = 1/(L*IN*OUT)
// Signs/scales folded into the coefficient matrix A.
// ============================================================================
__global__ __launch_bounds__(256) void k_idft(const float* __restrict__ ws,
                                              float* __restrict__ out) {
  __shared__ float coef[64][129];     // [o][k], padded
  int bx = blockIdx.x;                // (b*Hn+h)*8 + slice
  int slice = bx & 7, bh = bx >> 3;
  int tid = threadIdx.x, lane = tid & 31, wv = tid >> 5;
  const float* UR = ws + OFF_UR + (size_t)bh * 4096;
  const float* UI = ws + OFF_UI + (size_t)bh * 4096;
  const float sc = 1.0f / (2048.0f * 512.0f * 512.0f);

  for (int i = tid; i < 8192; i += 256) {
    int o = i >> 7, kk = i & 127, f = kk & 63;
    float v;
    if (kk < 64) v = sc * ((f == 0) ? 1.0f : 2.0f) * UR[o * 64 + f];
    else         v = (f == 0) ? 0.0f : (-2.0f * sc * UI[o * 64 + f]);
    coef[o][kk] = v;
  }
  __syncthreads();

  const float* cosb = ws + OFF_COS;
  const float* sinb = ws + OFF_SIN;
  int col0base = slice * 256 + wv * 32;
  for (int nt = 0; nt < 2; ++nt) {
    int col0 = col0base + nt * 16;
#pragma unroll
    for (int ot = 0; ot < 4; ++ot) {
      v8f acc = {};
      for (int k0 = 0; k0 < 128; k0 += 4) {
        v2f a = load_a(&coef[ot * 16][k0], 129, lane, 1.0f);
        const float* tab = (k0 < 64) ? (cosb + k0 * Ln) : (sinb + (k0 - 64) * Ln);
        int n = lane & 15, kk = (lane >> 4) << 1;
        v2f bb; bb.x = tab[kk * Ln + col0 + n]; bb.y = tab[(kk + 1) * Ln + col0 + n];
        acc = wmma4(a, bb, acc);
      }
      int n = lane & 15, mh = (lane >> 4) * 8;
      for (int r = 0; r < 8; ++r)
        out[((size_t)bh * 64 + ot * 16 + mh + r) * (size_t)Ln + col0 + n] = acc[r];
    }
  }
}

// ============================================================================
extern "C" void kernel_launch(void* const* d_in, const int* in_sizes, int n_in,
                              void* d_out, int out_size, void* d_ws, size_t ws_size,
                              hipStream_t stream) {
  (void)in_sizes; (void)n_in; (void)out_size; (void)ws_size;
  const float* q    = (const float*)d_in[0];
  const float* k    = (const float*)d_in[1];
  // d_in[2] = v : projected-but-unused in the reference -> skipped entirely
  const float* wq_w = (const float*)d_in[3];
  const float* wq_b = (const float*)d_in[4];
  const float* wk_w = (const float*)d_in[5];
  const float* wk_b = (const float*)d_in[6];
  // d_in[7,8] = wv_w/wv_b unused; d_in[11,12] = index arrays == arange(64)
  const float* w1   = (const float*)d_in[9];
  const float* w2   = (const float*)d_in[10];
  float* out = (float*)d_out;
  float* ws  = (float*)d_ws;

  k_basis   <<<(NF * Ln + 255) / 256, 256, 0, stream>>>(ws);
  k_dft_proj<<<Bn * Hn * 2,           256, 0, stream>>>(q, k, wq_w, wq_b, wk_w, wk_b, ws);
  k_attn    <<<Bn * Hn,               256, 0, stream>>>(ws);
  k_modemix <<<Hn * 64,               256, 0, stream>>>(w1, w2, ws);
  k_idft    <<<Bn * Hn * 8,           256, 0, stream>>>(ws, out);
}